// AdaptiveSlicing_23570780520663
// MI455X (gfx1250) — compile-verified
//
#include <hip/hip_runtime.h>
#include <hip/hip_bf16.h>

// ---------------------------------------------------------------------------
// AdaptiveSlicing fused attention-pooling for MI455X (gfx1250, wave32).
//
// Pipeline:
//   k_zero : zero global numer/denom accumulators in workspace
//   k_prep : K_c = slice_centers @ W_k + b_k ; inv_scale = 1/(4*clip(exp(lt)))
//   k_main : fused per-node pass (Q/V projection, logits, softmax,
//            attn_weights output, segment-sum via WMMA partial GEMMs)
//   k_tok  : tokens = LN((numer/denom) @ W_o + b_o), token_batch
//
// k_main's GEMM work uses V_WMMA_F32_16X16X4_F32 (exact fp32).
// ---------------------------------------------------------------------------

typedef float v2f __attribute__((ext_vector_type(2)));
typedef float v8f __attribute__((ext_vector_type(8)));

#define NSLICES 32
#define NHEADS  4
#define HD      16
#define BGR     16
#define NSLOT   3          // max distinct batch ids per 512-node block
#define WPB     8          // waves per block
#define TPW     4          // 16-node tiles per wave
#define NODES_PER_BLOCK (WPB * TPW * 16)   // 512

static __device__ __forceinline__ v8f wmma4(v2f a, v2f b, v8f c) {
  // D(16x16,f32) = A(16x4,f32) * B(4x16,f32) + C
  return __builtin_amdgcn_wmma_f32_16x16x4_f32(
      /*neg_a=*/false, a, /*neg_b=*/false, b,
      /*c_mod=*/(short)0, c, /*reuse_a=*/false, /*reuse_b=*/false);
}

// ---------------------------------------------------------------------------
__global__ void k_zero(float* p, int n) {
  int i = blockIdx.x * blockDim.x + threadIdx.x;
  if (i < n) p[i] = 0.0f;
}

// ---------------------------------------------------------------------------
__global__ void k_prep(const float* __restrict__ sc, const float* __restrict__ Wk,
                       const float* __restrict__ bk, const float* __restrict__ logT,
                       float* __restrict__ KcG, float* __restrict__ scal) {
  int tid = threadIdx.x;
  for (int o = tid; o < NSLICES * 64; o += blockDim.x) {
    int r = o >> 6, c = o & 63;
    float acc = bk[c];
    for (int e = 0; e < 64; ++e) acc += sc[r * 64 + e] * Wk[e * 64 + c];
    KcG[o] = acc;
  }
  if (tid == 0) {
    float t = __expf(logT[0]);
    t = fminf(fmaxf(t, 0.01f), 10.0f);
    scal[0] = 1.0f / (4.0f * t);   // 1/(sqrt(hd)*temp), hd=16
  }
}

// ---------------------------------------------------------------------------
__global__ __launch_bounds__(256)
void k_main(const float* __restrict__ z, const int* __restrict__ batch,
            const float* __restrict__ Wq, const float* __restrict__ Wv,
            const float* __restrict__ bq, const float* __restrict__ bv,
            const float* __restrict__ KcG, const float* __restrict__ scal,
            float* __restrict__ gNum, float* __restrict__ gDen,
            float* __restrict__ attnW, int N) {
  __shared__ __align__(16) float sWq[64][64];
  __shared__ __align__(16) float sWv[64][64];
  __shared__ __align__(16) float sKc[NSLICES][66];         // padded rows
  __shared__ __align__(16) float sBq[64];
  __shared__ __align__(16) float sBv[64];
  __shared__ __align__(16) float sQh[WPB][16][18];         // per-wave Q-head tile
  __shared__ __align__(16) float sVh[WPB][16][18];         // per-wave V-head tile
  __shared__ __align__(16) float sAttn[WPB][16][NSLICES];  // per-wave masked attn
  __shared__ __align__(16) float accNum[NSLOT][NHEADS][NSLICES][HD];
  __shared__ __align__(16) float accDen[NSLOT][NHEADS][NSLICES];
  __shared__ int sUsed[NSLOT];

  const int tid  = threadIdx.x;
  const int wave = tid >> 5;
  const int lane = tid & 31;
  const int half = lane >> 4;
  const int lh   = lane & 15;

  // ---- block init: weights to LDS, zero accumulators ----
  {
    const float4* Wq4 = (const float4*)Wq;
    const float4* Wv4 = (const float4*)Wv;
    float4* sWq4 = (float4*)&sWq[0][0];
    float4* sWv4 = (float4*)&sWv[0][0];
    for (int i = tid; i < 1024; i += 256) { sWq4[i] = Wq4[i]; sWv4[i] = Wv4[i]; }
    for (int i = tid; i < NSLICES * 64; i += 256) sKc[i >> 6][i & 63] = KcG[i];
    if (tid < 64) { sBq[tid] = bq[tid]; sBv[tid] = bv[tid]; }
    float* nf = &accNum[0][0][0][0];
    for (int i = tid; i < NSLOT * NHEADS * NSLICES * HD; i += 256) nf[i] = 0.0f;
    float* df = &accDen[0][0][0];
    for (int i = tid; i < NSLOT * NHEADS * NSLICES; i += 256) df[i] = 0.0f;
    if (tid < NSLOT) sUsed[tid] = 0;
  }
  const float inv_scale = scal[0];
  const int first_b = batch[blockIdx.x * NODES_PER_BLOCK];
  __syncthreads();

#pragma unroll 1
  for (int t = 0; t < TPW; ++t) {
    const int base  = blockIdx.x * NODES_PER_BLOCK + wave * (TPW * 16) + t * 16;
    const int myrow = base + lh;
    const int rowc  = (myrow < N) ? myrow : (N - 1);

    // A fragments of z tile (rows on lanes, K split across lane halves).
    // Reused for both the W_q and W_v GEMMs.
    const float* zr = z + (long long)rowc * 64 + 2 * half;
    v2f za[16];
#pragma unroll
    for (int c = 0; c < 16; ++c) za[c] = *(const v2f*)(zr + 4 * c);

    // slot id of my node (slot = batch - first_b, clamped; -1 = padded row)
    int slot;
    {
      int b = batch[rowc];
      slot = b - first_b;
      slot = slot < 0 ? 0 : (slot > NSLOT - 1 ? NSLOT - 1 : slot);
      if (myrow >= N) slot = -1;
    }
    int rowslot[8];
#pragma unroll
    for (int r = 0; r < 8; ++r) rowslot[r] = __shfl(slot, r + 8 * half);
    bool pres[NSLOT];
#pragma unroll
    for (int s = 0; s < NSLOT; ++s) pres[s] = (__any(slot == s) != 0);

    float aw0[8], aw1[8];
#pragma unroll
    for (int r = 0; r < 8; ++r) { aw0[r] = 0.0f; aw1[r] = 0.0f; }

#pragma unroll 1
    for (int h = 0; h < NHEADS; ++h) {
      // ---- Q_h and V_h projections: 16x16 outputs, K=64 ----
      v8f q  = {0, 0, 0, 0, 0, 0, 0, 0};
      v8f vv = {0, 0, 0, 0, 0, 0, 0, 0};
#pragma unroll
      for (int c = 0; c < 16; ++c) {
        v2f bfq, bfv;
        bfq.x = sWq[4 * c + 2 * half][16 * h + lh];
        bfq.y = sWq[4 * c + 2 * half + 1][16 * h + lh];
        bfv.x = sWv[4 * c + 2 * half][16 * h + lh];
        bfv.y = sWv[4 * c + 2 * half + 1][16 * h + lh];
        q  = wmma4(za[c], bfq, q);
        vv = wmma4(za[c], bfv, vv);
      }
      const float bqv = sBq[16 * h + lh];
      const float bvv = sBv[16 * h + lh];
#pragma unroll
      for (int r = 0; r < 8; ++r) {
        sQh[wave][r + 8 * half][lh] = (q[r] + bqv) * inv_scale;  // pre-scaled Q
        sVh[wave][r + 8 * half][lh] = vv[r] + bvv;
      }
      asm volatile("s_wait_dscnt 0x0" ::: "memory");

      // ---- logits: [16 nodes x 32 slices], K=16 (head dim) ----
      v8f l0 = {0, 0, 0, 0, 0, 0, 0, 0};
      v8f l1 = {0, 0, 0, 0, 0, 0, 0, 0};
#pragma unroll
      for (int c = 0; c < 4; ++c) {
        v2f a  = *(const v2f*)&sQh[wave][lh][4 * c + 2 * half];
        v2f b0 = *(const v2f*)&sKc[lh][16 * h + 4 * c + 2 * half];
        v2f b1 = *(const v2f*)&sKc[16 + lh][16 * h + 4 * c + 2 * half];
        l0 = wmma4(a, b0, l0);
        l1 = wmma4(a, b1, l1);
      }

      // ---- softmax over 32 slices (per node row, 16-lane groups) ----
      float a0[8], a1[8];
#pragma unroll
      for (int r = 0; r < 8; ++r) {
        float m = fmaxf(l0[r], l1[r]);
#pragma unroll
        for (int d = 1; d <= 8; d <<= 1) m = fmaxf(m, __shfl_xor(m, d));
        float e0 = __expf(l0[r] - m);
        float e1 = __expf(l1[r] - m);
        float ssum = e0 + e1;
#pragma unroll
        for (int d = 1; d <= 8; d <<= 1) ssum += __shfl_xor(ssum, d);
        float inv = 1.0f / ssum;
        a0[r] = e0 * inv; a1[r] = e1 * inv;
        aw0[r] += a0[r];  aw1[r] += a1[r];
      }

      // ---- segment partial sums per batch-slot present in this tile ----
#pragma unroll 1
      for (int s = 0; s < NSLOT; ++s) {
        if (!pres[s]) continue;                // wave-uniform branch
        if (lane == 0) sUsed[s] = 1;
        float d0 = 0.0f, d1 = 0.0f;
#pragma unroll
        for (int r = 0; r < 8; ++r) {
          float m = (rowslot[r] == s) ? 1.0f : 0.0f;
          float am0 = a0[r] * m, am1 = a1[r] * m;
          sAttn[wave][r + 8 * half][lh]      = am0;
          sAttn[wave][r + 8 * half][16 + lh] = am1;
          d0 += am0; d1 += am1;
        }
        asm volatile("s_wait_dscnt 0x0" ::: "memory");
        d0 += __shfl_xor(d0, 16);
        d1 += __shfl_xor(d1, 16);
        if (lane < 16) {
          atomicAdd(&accDen[s][h][lh], d0);
          atomicAdd(&accDen[s][h][16 + lh], d1);
        }
        // numer partial: [32 slices x 16 hd] = attn^T(32x16) * V(16x16)
#pragma unroll
        for (int mj = 0; mj < 2; ++mj) {
          v8f nacc = {0, 0, 0, 0, 0, 0, 0, 0};
#pragma unroll
          for (int c = 0; c < 4; ++c) {
            v2f a, b;
            a.x = sAttn[wave][4 * c + 2 * half][16 * mj + lh];
            a.y = sAttn[wave][4 * c + 2 * half + 1][16 * mj + lh];
            b.x = sVh[wave][4 * c + 2 * half][lh];
            b.y = sVh[wave][4 * c + 2 * half + 1][lh];
            nacc = wmma4(a, b, nacc);
          }
#pragma unroll
          for (int r = 0; r < 8; ++r)
            atomicAdd(&accNum[s][h][16 * mj + r + 8 * half][lh], nacc[r]);
        }
      }
    } // heads

    // ---- attn_weights output: mean over heads ----
#pragma unroll
    for (int r = 0; r < 8; ++r) {
      int node = base + r + 8 * half;
      if (node < N) {
        attnW[node * NSLICES + lh]      = 0.25f * aw0[r];
        attnW[node * NSLICES + 16 + lh] = 0.25f * aw1[r];
      }
    }
  } // tiles

  // ---- flush block accumulators to global (sparse atomic adds) ----
  __syncthreads();
  for (int s = 0; s < NSLOT; ++s) {
    if (!sUsed[s]) continue;
    int bidx = first_b + s;
    if (bidx >= BGR) continue;
    const float* nf = &accNum[s][0][0][0];
    for (int i = tid; i < NHEADS * NSLICES * HD; i += 256) {
      float v = nf[i];
      if (v != 0.0f) atomicAdd(&gNum[bidx * NHEADS * NSLICES * HD + i], v);
    }
    const float* df = &accDen[s][0][0];
    for (int i = tid; i < NHEADS * NSLICES; i += 256) {
      float v = df[i];
      if (v != 0.0f) atomicAdd(&gDen[bidx * NHEADS * NSLICES + i], v);
    }
  }
}

// ---------------------------------------------------------------------------
__global__ void k_tok(const float* __restrict__ gNum, const float* __restrict__ gDen,
                      const float* __restrict__ Wo, const float* __restrict__ bo,
                      const float* __restrict__ lng, const float* __restrict__ lnb,
                      float* __restrict__ outTok, float* __restrict__ outTB) {
  __shared__ float x[64];
  __shared__ float red[64];
  const int row = blockIdx.x;           // 0..511 = b*32 + k
  const int b = row >> 5, k = row & 31;
  const int t = threadIdx.x;            // 0..63
  const int h = t >> 4, dd = t & 15;

  float den = fmaxf(gDen[(b * NHEADS + h) * NSLICES + k], 1e-6f);
  x[t] = gNum[((b * NHEADS + h) * NSLICES + k) * HD + dd] / den;
  __syncthreads();

  float y = bo[t];
  for (int e = 0; e < 64; ++e) y += x[e] * Wo[e * 64 + t];

  red[t] = y;
  __syncthreads();
  for (int off = 32; off > 0; off >>= 1) {
    if (t < off) red[t] += red[t + off];
    __syncthreads();
  }
  float mu = red[0] * (1.0f / 64.0f);
  __syncthreads();
  float dvv = y - mu;
  red[t] = dvv * dvv;
  __syncthreads();
  for (int off = 32; off > 0; off >>= 1) {
    if (t < off) red[t] += red[t + off];
    __syncthreads();
  }
  float var = red[0] * (1.0f / 64.0f);

  outTok[row * 64 + t] = dvv * rsqrtf(var + 1e-5f) * lng[t] + lnb[t];
  if (t == 0) outTB[row] = (float)b;
}

// ---------------------------------------------------------------------------
extern "C" void kernel_launch(void* const* d_in, const int* in_sizes, int n_in,
                              void* d_out, int out_size, void* d_ws, size_t ws_size,
                              hipStream_t stream) {
  const float* z    = (const float*)d_in[0];
  const int*   batch= (const int*)  d_in[1];
  const float* sc   = (const float*)d_in[2];
  const float* logT = (const float*)d_in[3];
  const float* Wq   = (const float*)d_in[4];
  const float* bq   = (const float*)d_in[5];
  const float* Wk   = (const float*)d_in[6];
  const float* bk   = (const float*)d_in[7];
  const float* Wv   = (const float*)d_in[8];
  const float* bv   = (const float*)d_in[9];
  const float* Wo   = (const float*)d_in[10];
  const float* bo   = (const float*)d_in[11];
  const float* lng  = (const float*)d_in[12];
  const float* lnb  = (const float*)d_in[13];

  const int N = in_sizes[0] / 64;

  float* ws   = (float*)d_ws;
  float* gNum = ws;                    // 16*4*32*16 = 32768
  float* gDen = ws + 32768;            // 16*4*32   = 2048
  float* KcG  = ws + 34816;            // 32*64     = 2048
  float* scal = ws + 36864;            // 1

  float* outTok = (float*)d_out;                 // [512,64]
  float* outTB  = outTok + 512 * 64;             // [512]
  float* attnW  = outTB + 512;                   // [N,32]

  k_zero<<<(34816 + 255) / 256, 256, 0, stream>>>(gNum, 34816);
  k_prep<<<1, 256, 0, stream>>>(sc, Wk, bk, logT, KcG, scal);

  const int nblocks = (N + NODES_PER_BLOCK - 1) / NODES_PER_BLOCK;
  k_main<<<nblocks, 256, 0, stream>>>(z, batch, Wq, Wv, bq, bv, KcG, scal,
                                      gNum, gDen, attnW, N);
  k_tok<<<512, 64, 0, stream>>>(gNum, gDen, Wo, bo, lng, lnb, outTok, outTB);
}